// DualSparseEncoder_34153579938157
// MI455X (gfx1250) — compile-verified
//
#include <hip/hip_runtime.h>
#include <math.h>

#define QN 32
#define DN 256
#define VOCABN 30522
#define FN 3
#define DSTRIDE (DN + 1)

typedef float v2f __attribute__((ext_vector_type(2)));
typedef float v8f __attribute__((ext_vector_type(8)));

__device__ __forceinline__ float wave_max(float v) {
#pragma unroll
  for (int off = 16; off; off >>= 1) v = fmaxf(v, __shfl_xor(v, off, 32));
  return v;
}
__device__ __forceinline__ float wave_sum(float v) {
#pragma unroll
  for (int off = 16; off; off >>= 1) v += __shfl_xor(v, off, 32);
  return v;
}

// One block per batch row b. 256 threads = 8 wave32 waves; thread t owns d = t.
__global__ __launch_bounds__(256)
void dse_score_kernel(const int* __restrict__ q_ids,
                      const float* __restrict__ q_w,
                      const int* __restrict__ d_ids,
                      const float* __restrict__ d_logits,
                      float* __restrict__ scores /* [B][FN] */) {
  __shared__ int   s_qid[QN];
  __shared__ float s_qw[QN];
  __shared__ int   s_did[DN];
  __shared__ float s_exact[QN][DSTRIDE];
  __shared__ float s_part[QN * 8];
  __shared__ float s_acc1[8];
  __shared__ float s_acc2[8];
  __shared__ float s_score0;

  const int b    = blockIdx.x;
  const int tid  = (int)threadIdx.x;
  const int lane = tid & 31;
  const int wave = tid >> 5;

  if (tid < QN) { s_qid[tid] = q_ids[b * QN + tid]; s_qw[tid] = q_w[b * QN + tid]; }
  s_did[tid] = d_ids[b * DN + tid];
  __syncthreads();

  // Gather phase: 32 independent gathers per thread (latency-bound; maximize ILP).
  const float* row = d_logits + ((long long)b * DN + tid) * (long long)VOCABN;
  float tr[QN];
#pragma unroll
  for (int q = 0; q < QN; ++q) tr[q] = s_qw[q] * row[s_qid[q]];

  const int myDid = s_did[tid];
#pragma unroll
  for (int q = 0; q < QN; ++q)
    s_exact[q][tid] = (s_qid[q] == myDid) ? tr[q] : 0.0f;

  // score0: per-q max over d (can be negative -> true max), then sum over q.
#pragma unroll
  for (int q = 0; q < QN; ++q) {
    float m = wave_max(tr[q]);            // max over this wave's 32 d values
    if (lane == 0) s_part[q * 8 + wave] = m;
  }
  __syncthreads();

  if (wave == 0) {
    float m = s_part[lane * 8];           // lane == q
#pragma unroll
    for (int w = 1; w < 8; ++w) m = fmaxf(m, s_part[lane * 8 + w]);
    float s = wave_sum(m);
    if (lane == 0) s_score0 = s;
  }

  // score1: ordered diagonal 3-gram windows on exact. Values are sums of
  // strictly-positive entries when masked-in, else 0, so init-0 max matches ref.
  float acc1 = 0.0f;
  for (int q = wave; q < QN - 2; q += 8) {
    float m = 0.0f;
    for (int d = lane; d < DN - 2; d += 32) {
      float e0 = s_exact[q][d];
      float e1 = s_exact[q + 1][d + 1];
      float e2 = s_exact[q + 2][d + 2];
      if (e0 > 0.0f && e1 > 0.0f && e2 > 0.0f) m = fmaxf(m, e0 + e1 + e2);
    }
    acc1 += wave_max(m);
  }

  // score2: proximity = sliding max-of-8 over d, then 3-window over q.
  float acc2 = 0.0f;
  for (int q = wave; q < QN - 2; q += 8) {
    float m = 0.0f;
    for (int d = lane; d < DN - 7; d += 32) {
      float p0 = s_exact[q][d], p1 = s_exact[q + 1][d], p2 = s_exact[q + 2][d];
#pragma unroll
      for (int i = 1; i < 8; ++i) {
        p0 = fmaxf(p0, s_exact[q][d + i]);
        p1 = fmaxf(p1, s_exact[q + 1][d + i]);
        p2 = fmaxf(p2, s_exact[q + 2][d + i]);
      }
      if (p0 > 0.0f && p1 > 0.0f && p2 > 0.0f) m = fmaxf(m, p0 + p1 + p2);
    }
    acc2 += wave_max(m);
  }
  if (lane == 0) { s_acc1[wave] = acc1; s_acc2[wave] = acc2; }
  __syncthreads();

  if (tid == 0) {
    float s1 = 0.0f, s2 = 0.0f;
#pragma unroll
    for (int w = 0; w < 8; ++w) { s1 += s_acc1[w]; s2 += s_acc2[w]; }
    scores[b * FN + 0] = s_score0;
    scores[b * FN + 1] = s1;
    scores[b * FN + 2] = s2;
  }
}

// Single wave32: ragged segment-max over passage groups, then the [P,FN]@[FN]
// linear head done as one V_WMMA_F32_16X16X4_F32 (A 16x4 zero-padded, B 4x16
// broadcast weights, D[p][0] read from col-0 lanes).
__global__ __launch_bounds__(32)
void dse_finish_kernel(const float* __restrict__ scores,
                       const int* __restrict__ psg_offset,
                       const float* __restrict__ lin_w,
                       const float* __restrict__ lin_b,
                       float* __restrict__ out, int P) {
  __shared__ float sA[16 * 4];
  __shared__ float sw[4];
  const int tid = (int)threadIdx.x;
  sA[tid] = 0.0f;
  sA[tid + 32] = 0.0f;
  if (tid < 4) sw[tid] = (tid < FN) ? lin_w[tid] : 0.0f;
  __syncthreads();
  for (int idx = tid; idx < P * FN; idx += 32) {
    int p = idx / FN, f = idx % FN;
    int b0 = psg_offset[p], b1 = psg_offset[p + 1];
    float m = -__builtin_inff();
    for (int bb = b0; bb < b1; ++bb) m = fmaxf(m, scores[bb * FN + f]);
    sA[p * 4 + f] = m;
  }
  __syncthreads();
  const float bias = lin_b[0];
#if __has_builtin(__builtin_amdgcn_wmma_f32_16x16x4_f32)
  // 32-bit A 16x4 layout: VGPR v, lane L -> A[L&15][v + 2*(L>>4)].
  // B 4x16 (row striped across lanes): VGPR v, lane L -> K = v + 2*(L>>4).
  const int half = tid >> 4, l = tid & 15;
  v2f a, bm;
  a[0]  = sA[l * 4 + (2 * half + 0)];
  a[1]  = sA[l * 4 + (2 * half + 1)];
  bm[0] = sw[2 * half + 0];
  bm[1] = sw[2 * half + 1];
  v8f c = {};
  v8f dm = __builtin_amdgcn_wmma_f32_16x16x4_f32(
      /*neg_a=*/false, a, /*neg_b=*/false, bm,
      /*c_mod=*/(short)0, c, /*reuse_a=*/false, /*reuse_b=*/false);
  // D layout: VGPR v, lane L -> row (v + 8*(L>>4)), col (L&15). Col 0 holds out.
  if (l == 0) {
#pragma unroll
    for (int v = 0; v < 8; ++v) {
      int p = v + 8 * half;
      if (p < P) out[p] = dm[v] + bias;
    }
  }
#else
  if (tid < P) {
    float s = bias;
    for (int f = 0; f < FN; ++f) s += sA[tid * 4 + f] * sw[f];
    out[tid] = s;
  }
#endif
}

extern "C" void kernel_launch(void* const* d_in, const int* in_sizes, int n_in,
                              void* d_out, int out_size, void* d_ws, size_t ws_size,
                              hipStream_t stream) {
  (void)n_in; (void)out_size; (void)ws_size;
  const int*   q_ids   = (const int*)  d_in[0];
  const float* q_w     = (const float*)d_in[1];
  const int*   d_ids   = (const int*)  d_in[2];
  const float* d_log   = (const float*)d_in[3];
  const int*   psg_off = (const int*)  d_in[4];
  const float* lin_w   = (const float*)d_in[5];
  const float* lin_b   = (const float*)d_in[6];

  const int B = in_sizes[0] / QN;   // 16
  const int P = in_sizes[4] - 1;    // 8
  float* scores = (float*)d_ws;     // [B][FN] scratch, fully overwritten each call

  dse_score_kernel<<<B, 256, 0, stream>>>(q_ids, q_w, d_ids, d_log, scores);
  dse_finish_kernel<<<1, 32, 0, stream>>>(scores, psg_off, lin_w, lin_b,
                                          (float*)d_out, P);
}